// CgTransform_41583873360342
// MI455X (gfx1250) — compile-verified
//
#include <hip/hip_runtime.h>

typedef __attribute__((ext_vector_type(2))) float v2f;
typedef __attribute__((ext_vector_type(8))) float v8f;

// ---------------- compile-time problem constants (L=7, T=128) ----------------
constexpr int LMAX = 7;
constexpr int LS   = 64;            // (L+1)^2
constexpr int Tn   = 128;
constexpr long long TT = (long long)Tn * Tn;   // 16384

constexpr int countNPK() {
  int n = 0;
  for (int a = 0; a <= LMAX; ++a)
    for (int b = 0; b <= LMAX; ++b) {
      int d = a > b ? a - b : b - a;
      int s = (a + b) < LMAX ? (a + b) : LMAX;
      n += (s + 1) * (s + 1) - d * d;
    }
  return n;
}
constexpr int NPK = countNPK();     // 2416 (pair, k-row) blocks

// packed per-block geometry + 64-bit flat output base of the 128x128 block
struct Blk {
  int info[NPK];        // l1s | l2s<<6 | J<<12 | S<<16 | kg<<19 | I<<25
  long long base[NPK];
};

constexpr Blk buildBlk() {
  Blk B{};
  int np[8] = {};
  for (int l = 0; l <= LMAX; ++l) {
    int c = 0;
    for (int a = 0; a <= LMAX; ++a)
      for (int b = 0; b <= LMAX; ++b) {
        int d = a > b ? a - b : b - a;
        int s = (a + b) < LMAX ? (a + b) : LMAX;
        if (d <= l && l <= s) ++c;
      }
    np[l] = c;
  }
  long long outBase[8] = {};
  long long rowPitch[8] = {};
  long long base = 0;
  for (int l = 0; l <= LMAX; ++l) {
    outBase[l]  = base;
    rowPitch[l] = (long long)np[l] * TT;
    base += (long long)(2 * l + 1) * rowPitch[l];
  }
  int qc[8] = {};
  int n = 0;
  for (int a = 0; a <= LMAX; ++a)
    for (int b = 0; b <= LMAX; ++b) {
      const int d = a > b ? a - b : b - a;
      const int s = (a + b) < LMAX ? (a + b) : LMAX;
      const int l1s = a * a, l2s = b * b;
      const int I = 2 * a + 1, J = 2 * b + 1;
      const int S = (I + 3) / 4;
      long long co[8] = {};
      for (int l = d; l <= s; ++l) { co[l] = (long long)qc[l] * TT; ++qc[l]; }
      const int K = (s + 1) * (s + 1) - d * d;
      for (int kl = 0; kl < K; ++kl) {
        const int kg = d * d + kl;
        int l = 0;
        while ((l + 1) * (l + 1) <= kg) ++l;
        const int m = kg - l * l;
        B.info[n] = l1s | (l2s << 6) | (J << 12) | (S << 16) | (kg << 19) | (I << 25);
        B.base[n] = outBase[l] + (long long)m * rowPitch[l] + co[l];
        ++n;
      }
    }
  return B;
}

__constant__ Blk kBlk = buildBlk();

constexpr int PADB = 132;   // sB row stride (floats) — breaks bank conflicts
constexpr int PD2  = 132;   // sD pair-row stride (v2f units)

// -------------------- S-specialized WMMA main loop --------------------------
template <int S>
__device__ __forceinline__ void cg_main(const float* __restrict__ clms,
                                        float* __restrict__ out,
                                        const v2f* sD, long long base, int l1s,
                                        int wave, int laneLo, int laneHi) {
  // A fragments (clms_l1 columns for this wave's t1-tile), loaded once.
  // Rows beyond I multiply against D==0 so only an address clamp is needed.
  v2f af[S];
  const int t1 = wave * 16 + laneLo;
#pragma unroll
  for (int s = 0; s < S; ++s) {
    int r0 = l1s + 4 * s + 2 * laneHi;
    int r1 = r0 + 1;
    r0 = r0 < 63 ? r0 : 63;
    r1 = r1 < 63 ? r1 : 63;
    af[s] = (v2f){clms[r0 * Tn + t1], clms[r1 * Tn + t1]};
  }

#pragma unroll
  for (int nt = 0; nt < 8; ++nt) {
    v8f c = (v8f){0, 0, 0, 0, 0, 0, 0, 0};
#pragma unroll
    for (int s = 0; s < S; ++s) {
      const int i2 = 2 * s + laneHi;                 // pair-row of D
      const v2f bf = sD[i2 * PD2 + nt * 16 + laneLo]; // one ds_load_b64
      c = __builtin_amdgcn_wmma_f32_16x16x4_f32(false, af[s], false, bf,
                                                (short)0, c, false, false);
    }
    const int col = nt * 16 + laneLo;
#pragma unroll
    for (int r = 0; r < 8; ++r) {
      const int t1r = wave * 16 + r + 8 * laneHi;    // C/D tile rows r / r+8
      out[base + t1r * Tn + col] = c[r];
    }
  }
}

// ---------------------------------- kernel ----------------------------------
__global__ __launch_bounds__(256) void cg_wmma_kernel(const float* __restrict__ clms,
                                                      const float* __restrict__ Cg,
                                                      float* __restrict__ out) {
  __shared__ float sB[15 * PADB];     // clms_l2 rows [j][t2]
  __shared__ float sCgRow[16 * 16];   // CG slice C[kg, l1s+i, l2s+j], zero-padded
  __shared__ v2f   sD[8 * PD2];       // D pair-interleaved: {D[2i2], D[2i2+1]}[t2]

  const int n   = blockIdx.x;
  const int tid = threadIdx.x;

  const int info = kBlk.info[n];
  const long long base = kBlk.base[n];
  const int l1s = info & 63;
  const int l2s = (info >> 6) & 63;
  const int J   = (info >> 12) & 15;
  const int S   = (info >> 16) & 7;
  const int kg  = (info >> 19) & 63;
  const int I   = (info >> 25) & 15;

  // ---- stage clms_l2 rows ----
  for (int idx = tid; idx < J * Tn; idx += 256) {
    const int j  = idx >> 7;
    const int t2 = idx & 127;
    sB[j * PADB + t2] = clms[(l2s + j) * Tn + t2];
  }
  // ---- stage the 16x16 CG slice for row kg (zero-padded) ----
  {
    const int i = tid >> 4, j = tid & 15;            // blockDim == 256 == 16*16
    float v = 0.0f;
    if (i < I && j < J) v = Cg[kg * 4096 + (l1s + i) * LS + (l2s + j)];
    sCgRow[tid] = v;
  }
  __syncthreads();

  // ---- stage 1: D[i, t2] = sum_j CG[i,j] * clms_l2[j, t2]  (i padded to 16) --
  for (int idx = tid; idx < 16 * Tn; idx += 256) {
    const int i  = idx >> 7;                         // wave-uniform
    const int t2 = idx & 127;
    float acc = 0.0f;
    if (i < I) {
      for (int j = 0; j < J; ++j)
        acc = fmaf(sCgRow[i * 16 + j], sB[j * PADB + t2], acc);
    }
    ((float*)&sD[(i >> 1) * PD2 + t2])[i & 1] = acc;
  }
  __syncthreads();

  // ---- stage 2: out[t1, t2] = clms_l1^T x D via WMMA ----
  const int lane   = tid & 31;
  const int wave   = tid >> 5;        // t1-tile 0..7
  const int laneLo = lane & 15;
  const int laneHi = lane >> 4;

  switch (S) {  // uniform -> one scalar branch; EXEC all-1s across WMMAs
    case 1: cg_main<1>(clms, out, sD, base, l1s, wave, laneLo, laneHi); break;
    case 2: cg_main<2>(clms, out, sD, base, l1s, wave, laneLo, laneHi); break;
    case 3: cg_main<3>(clms, out, sD, base, l1s, wave, laneLo, laneHi); break;
    default: cg_main<4>(clms, out, sD, base, l1s, wave, laneLo, laneHi); break;
  }
}

// --------------------------------- launcher ---------------------------------
extern "C" void kernel_launch(void* const* d_in, const int* in_sizes, int n_in,
                              void* d_out, int out_size, void* d_ws, size_t ws_size,
                              hipStream_t stream) {
  (void)in_sizes; (void)n_in; (void)out_size; (void)d_ws; (void)ws_size;
  const float* clms = (const float*)d_in[0];  // (64, 128) f32
  const float* Cg   = (const float*)d_in[1];  // (64, 64, 64) f32
  float* out        = (float*)d_out;          // 2416 * 16384 f32

  cg_wmma_kernel<<<dim3(NPK), 256, 0, stream>>>(clms, Cg, out);
}